// DifferentiableKF_41618233098293
// MI455X (gfx1250) — compile-verified
//
#include <hip/hip_runtime.h>

typedef __attribute__((ext_vector_type(2))) float v2f;
typedef __attribute__((ext_vector_type(8))) float v8f;
typedef __attribute__((ext_vector_type(4))) unsigned int u32x4;
typedef __attribute__((ext_vector_type(8))) int i32x8;
typedef __attribute__((ext_vector_type(4))) int i32x4;

#define NN 128
#define LS 132                 // padded LDS row stride in floats (conflict-free)
#define MAT_F (NN * LS)        // floats per LDS matrix buffer
#define PB 18                  // 16x16 pivot-block scratch row stride (even -> v2f aligned)

// ---------------- Tensor Data Mover: load 128x128 f32 tile -> LDS -----------------
// D# pad feature inserts 4 DWORDs after every 128 DWORDs => LDS row stride 132 floats.
// Dynamic-LDS base offset is 0 in this kernel (no static __shared__), so lds_byte_off
// is the raw offset of our buffer.
__device__ __forceinline__ void tdm_load_mat128(const float* __restrict__ gsrc,
                                                unsigned lds_byte_off) {
  const unsigned long long ga = (unsigned long long)(size_t)gsrc;
  u32x4 g0;
  g0[0] = 1u;                                            // count=1, user descriptor
  g0[1] = lds_byte_off;                                  // lds_addr [63:32]
  g0[2] = (unsigned)(ga & 0xFFFFFFFFu);                  // global_addr [95:64]
  g0[3] = (unsigned)((ga >> 32) & 0x1FFFFFFu)            // global_addr [120:96]
        | (2u << 30);                                    // type=2 (image)
  i32x8 g1;
  g1[0] = (int)((2u << 16)      // data_size = 4B
              | (1u << 20)      // pad_enable
              | (6u << 22)      // pad_interval code 6 = 128 DWORDs
              | (3u << 25));    // pad_amount  code 3 = 4 DWORDs
  g1[1] = (int)(128u << 16);    // tensor_dim0 low16 (bits 63:48) = 128
  g1[2] = (int)(128u << 16);    // tensor_dim0 hi=0 | tensor_dim1 low16 = 128
  g1[3] = (int)(128u << 16);    // tensor_dim1 hi=0 | tile_dim0 = 128
  g1[4] = 128;                  // tile_dim1 = 128, tile_dim2 = 0
  g1[5] = 128;                  // tensor_dim0_stride low32 = 128
  g1[6] = 0;                    // stride hi | tensor_dim1_stride (unused, 2D tile)
  g1[7] = 0;
  const i32x4 zz4 = {0, 0, 0, 0};
  const i32x8 zz8 = {0, 0, 0, 0, 0, 0, 0, 0};
  // clang-23 / therock-10.0 form: 6 args (g0, g1, g2, g3, extra group, cpol)
  __builtin_amdgcn_tensor_load_to_lds(g0, g1, zz4, zz4, zz8, 0);
}

// ---- WMMA helpers: each wave computes one 16-row block (8 tiles of 16x16) ----

// C += A_rowmajor(LDS) * B_rowmajor(LDS)
__device__ __forceinline__ void wmma_rowblock_rm(const float* __restrict__ Ab,
                                                 const float* __restrict__ Bb,
                                                 int m0, int lh, int ll, v8f acc[8]) {
  for (int k0 = 0; k0 < NN; k0 += 4) {
    v2f a = *(const v2f*)(Ab + (m0 + ll) * LS + k0 + 2 * lh);   // A 16x4 frag
    const int k = k0 + 2 * lh;
#pragma unroll
    for (int nb = 0; nb < 8; ++nb) {
      const int n = nb * 16 + ll;
      v2f b;
      b.x = Bb[k * LS + n];
      b.y = Bb[(k + 1) * LS + n];
      acc[nb] = __builtin_amdgcn_wmma_f32_16x16x4_f32(
          false, a, false, b, (short)0, acc[nb], false, false);
    }
  }
}

// C += A_rowmajor(LDS) * (Bt)^T  where Bt is row-major in LDS: B[k][n] = Bt[n][k]
__device__ __forceinline__ void wmma_rowblock_bt(const float* __restrict__ Ab,
                                                 const float* __restrict__ Bt,
                                                 int m0, int lh, int ll, v8f acc[8]) {
  for (int k0 = 0; k0 < NN; k0 += 4) {
    v2f a = *(const v2f*)(Ab + (m0 + ll) * LS + k0 + 2 * lh);
    const int k = k0 + 2 * lh;
#pragma unroll
    for (int nb = 0; nb < 8; ++nb) {
      const int n = nb * 16 + ll;
      v2f b = *(const v2f*)(Bt + n * LS + k);   // contiguous: transpose for free
      acc[nb] = __builtin_amdgcn_wmma_f32_16x16x4_f32(
          false, a, false, b, (short)0, acc[nb], false, false);
    }
  }
}

__device__ __forceinline__ void store_acc_lds(float* __restrict__ Out,
                                              int m0, int lh, int ll, const v8f acc[8]) {
#pragma unroll
  for (int nb = 0; nb < 8; ++nb)
#pragma unroll
    for (int r = 0; r < 8; ++r)
      Out[(m0 + r + 8 * lh) * LS + nb * 16 + ll] = acc[nb][r];
}

__global__ __launch_bounds__(256) void
kf_update_kernel(const float* __restrict__ z_prev, const float* __restrict__ P_prev,
                 const float* __restrict__ A,      const float* __restrict__ Q,
                 const float* __restrict__ a_obs,  const float* __restrict__ R_obs,
                 float* __restrict__ z_out,        float* __restrict__ P_out) {
  extern __shared__ float smem[];
  float* buf0 = smem;                // P_prev -> P_pred
  float* buf1 = smem + MAT_F;        // T -> S -> inv(S)
  float* buf2 = smem + 2 * MAT_F;    // A -> K
  float* zv   = smem + 3 * MAT_F;    // z_prev        [128]
  float* zp   = zv + NN;             // z_pred        [128]
  float* inn  = zp + NN;             // a_obs - z_pred[128]
  float* Pblk = inn + NN;            // 16x16 pivot-block inverse, stride PB [16*PB]
  float* rowW = Pblk + 16 * PB;      // W = P @ S[k][:] row block [16*LS]

  const int b    = blockIdx.x;
  const int tid  = threadIdx.x;
  const int lane = tid & 31;
  const int ll   = lane & 15;
  const int lh   = lane >> 4;
  const int w    = tid >> 5;                 // wave id 0..7
  const int m0   = w * 16;                   // wave's output row block
  const size_t mb = (size_t)b * (NN * NN);

  // ---- stage A -> buf2 and P_prev -> buf0 via Tensor Data Mover (wave 0 issues) ----
  if (w == 0) {
    tdm_load_mat128(A, 2u * MAT_F * sizeof(float));     // A -> buf2
    tdm_load_mat128(P_prev + mb, 0u);                   // P_prev -> buf0
    __builtin_amdgcn_s_wait_tensorcnt(0);
  }
  if (tid < NN) zv[tid] = z_prev[(size_t)b * NN + tid];
  __syncthreads();

  // ---- z_pred = A @ z_prev ; innovation = a_obs - z_pred (waves 0-3) ----
  if (tid < NN) {
    float s = 0.f;
    for (int k = 0; k < NN; ++k) s = fmaf(buf2[tid * LS + k], zv[k], s);
    zp[tid]  = s;
    inn[tid] = a_obs[(size_t)b * NN + tid] - s;
  }

  v8f acc[8];
  const v8f vzero = {0.f, 0.f, 0.f, 0.f, 0.f, 0.f, 0.f, 0.f};

  // ---- Phase A: T = A @ P_prev -> buf1 ----
#pragma unroll
  for (int i = 0; i < 8; ++i) acc[i] = vzero;
  wmma_rowblock_rm(buf2, buf0, m0, lh, ll, acc);
  store_acc_lds(buf1, m0, lh, ll, acc);
  __syncthreads();

  // ---- Phase B: P_pred = T @ A^T + Q -> buf0 ----
#pragma unroll
  for (int nb = 0; nb < 8; ++nb)
#pragma unroll
    for (int r = 0; r < 8; ++r)
      acc[nb][r] = Q[mb + (size_t)(m0 + r + 8 * lh) * NN + nb * 16 + ll];
  wmma_rowblock_bt(buf1, buf2, m0, lh, ll, acc);
  store_acc_lds(buf0, m0, lh, ll, acc);
  __syncthreads();

  // ---- Phase C: S = P_pred + R_obs -> buf1 ----
  for (int idx = tid; idx < NN * NN; idx += 256) {
    const int row = idx >> 7, col = idx & 127;
    buf1[row * LS + col] = buf0[row * LS + col] + R_obs[mb + idx];
  }

  // ---- Blocked Gauss-Jordan inversion of S (SPD: no pivoting), 16x16 blocks ----
  // Trailing O(N^3) work runs on WMMA; only the 8 tiny 16x16 pivot inversions are
  // scalar, done wave-synchronously inside one wave (wave32 in-order LDS).
  const int pr = lane & 15;          // pivot-block row owned by this lane
  const int ph = (lane >> 4) * 8;    // pivot-block column half (8 cols)
  for (int k = 0; k < 8; ++k) {
    const int kbase = k * 16;
    __syncthreads();                 // prior iteration / Phase C writes visible

    if (w == k) {
      // copy pivot block S[k][k] -> Pblk and invert in place (scalar GJ, 1 wave)
      for (int c = 0; c < 8; ++c)
        Pblk[pr * PB + ph + c] = buf1[(kbase + pr) * LS + kbase + ph + c];
      __builtin_amdgcn_wave_barrier();
      for (int kk = 0; kk < 16; ++kk) {
        const float piv = Pblk[kk * PB + kk];
        const float rp  = 1.0f / piv;
        __builtin_amdgcn_wave_barrier();
        if (pr == kk) {
          for (int c = 0; c < 8; ++c) {
            const int cc = ph + c;
            const float v = (cc == kk) ? 1.0f : Pblk[kk * PB + cc];
            Pblk[kk * PB + cc] = v * rp;
          }
        }
        __builtin_amdgcn_wave_barrier();
        const float f = (pr == kk) ? 0.0f : Pblk[pr * PB + kk];
        __builtin_amdgcn_wave_barrier();
        if (pr != kk) {
          for (int c = 0; c < 8; ++c) {
            const int cc = ph + c;
            const float base = (cc == kk) ? 0.0f : Pblk[pr * PB + cc];
            Pblk[pr * PB + cc] = fmaf(-f, Pblk[kk * PB + cc], base);
          }
        }
        __builtin_amdgcn_wave_barrier();
      }
    }
    __syncthreads();                 // Pblk ready; S[k][:] still original

    // step 2: W = P @ S[k][:]  (each wave one 16x16 tile, WMMA K=16)
    {
      v8f t = vzero;
#pragma unroll
      for (int k0 = 0; k0 < 16; k0 += 4) {
        v2f a = *(const v2f*)(Pblk + ll * PB + k0 + 2 * lh);
        const int kk = k0 + 2 * lh;
        const int n = m0 + ll;
        v2f bfr;
        bfr.x = buf1[(kbase + kk) * LS + n];
        bfr.y = buf1[(kbase + kk + 1) * LS + n];
        t = __builtin_amdgcn_wmma_f32_16x16x4_f32(false, a, false, bfr,
                                                  (short)0, t, false, false);
      }
#pragma unroll
      for (int r = 0; r < 8; ++r)
        rowW[(r + 8 * lh) * LS + m0 + ll] = t[r];
    }
    __syncthreads();                 // W ready

    // step 3: wave k installs row block; others do rank-16 trailing update
    if (w == k) {
      for (int e = lane; e < 16 * NN; e += 32) {
        const int rr = e >> 7, cc = e & 127;
        buf1[(kbase + rr) * LS + cc] = rowW[rr * LS + cc];
      }
      __builtin_amdgcn_wave_barrier();
      for (int c = 0; c < 8; ++c)
        buf1[(kbase + pr) * LS + kbase + ph + c] = Pblk[pr * PB + ph + c];
    } else {
      const int ibase = w * 16;
      // save F = S[i][k] as A-operand fragments before overwriting the column block
      v2f fa[4];
#pragma unroll
      for (int k0 = 0; k0 < 16; k0 += 4)
        fa[k0 >> 2] = *(const v2f*)(buf1 + (ibase + ll) * LS + kbase + k0 + 2 * lh);
#pragma unroll
      for (int j = 0; j < 8; ++j) {
        v8f t = vzero;
        if (j == k) {
          // S[i][k] = -(F @ P)
#pragma unroll
          for (int k0 = 0; k0 < 16; k0 += 4) {
            const int kk = k0 + 2 * lh;
            v2f bfr;
            bfr.x = Pblk[kk * PB + ll];
            bfr.y = Pblk[(kk + 1) * PB + ll];
            t = __builtin_amdgcn_wmma_f32_16x16x4_f32(false, fa[k0 >> 2], false, bfr,
                                                      (short)0, t, false, false);
          }
#pragma unroll
          for (int r = 0; r < 8; ++r)
            buf1[(ibase + r + 8 * lh) * LS + kbase + ll] = -t[r];
        } else {
          // S[i][j] -= F @ W_j   (f32 WMMA has no A-negate: fold as RMW on D tile)
          const int n = j * 16 + ll;
#pragma unroll
          for (int k0 = 0; k0 < 16; k0 += 4) {
            const int kk = k0 + 2 * lh;
            v2f bfr;
            bfr.x = rowW[kk * LS + n];
            bfr.y = rowW[(kk + 1) * LS + n];
            t = __builtin_amdgcn_wmma_f32_16x16x4_f32(false, fa[k0 >> 2], false, bfr,
                                                      (short)0, t, false, false);
          }
#pragma unroll
          for (int r = 0; r < 8; ++r) {
            const int idx = (ibase + r + 8 * lh) * LS + j * 16 + ll;
            buf1[idx] = buf1[idx] - t[r];
          }
        }
      }
    }
  }
  __syncthreads();                   // inv(S) ready in buf1

  // ---- Phase D: K = P_pred @ inv(S) -> buf2 ----
#pragma unroll
  for (int i = 0; i < 8; ++i) acc[i] = vzero;
  wmma_rowblock_rm(buf0, buf1, m0, lh, ll, acc);
  store_acc_lds(buf2, m0, lh, ll, acc);
  __syncthreads();

  // ---- z_curr = z_pred + K @ innovation (waves 0-3) ----
  if (tid < NN) {
    float s = zp[tid];
    for (int k = 0; k < NN; ++k) s = fmaf(buf2[tid * LS + k], inn[k], s);
    z_out[(size_t)b * NN + tid] = s;
  }

  // ---- Phase E: P_curr = P_pred - K @ P_pred, streamed straight to HBM ----
#pragma unroll
  for (int i = 0; i < 8; ++i) acc[i] = vzero;
  wmma_rowblock_rm(buf2, buf0, m0, lh, ll, acc);
#pragma unroll
  for (int nb = 0; nb < 8; ++nb)
#pragma unroll
    for (int r = 0; r < 8; ++r) {
      const int row = m0 + r + 8 * lh;
      const int col = nb * 16 + ll;
      P_out[mb + (size_t)row * NN + col] = buf0[row * LS + col] - acc[nb][r];
    }
}

extern "C" void kernel_launch(void* const* d_in, const int* in_sizes, int n_in,
                              void* d_out, int out_size, void* d_ws, size_t ws_size,
                              hipStream_t stream) {
  const float* z_prev = (const float*)d_in[0];
  const float* P_prev = (const float*)d_in[1];
  const float* A      = (const float*)d_in[2];
  const float* Q      = (const float*)d_in[3];
  const float* a_obs  = (const float*)d_in[4];
  const float* R_obs  = (const float*)d_in[5];

  const int B = in_sizes[1] / (NN * NN);   // 2048
  float* z_out = (float*)d_out;                         // [B,128,1] first
  float* P_out = z_out + (size_t)B * NN;                // then [B,128,128]

  const size_t shmem =
      (size_t)(3 * MAT_F + 3 * NN + 16 * PB + 16 * LS) * sizeof(float); // ~214 KB < 320 KB/WGP
  kf_update_kernel<<<B, 256, shmem, stream>>>(z_prev, P_prev, A, Q, a_obs, R_obs,
                                              z_out, P_out);
}